// Angle_5119601016909
// MI455X (gfx1250) — compile-verified
//
#include <hip/hip_runtime.h>

// DimeNet spherical-basis layer for MI455X (gfx1250).
// Phase 1: rbf[E,42]  (spherical Bessel * envelope), coalesced elementwise;
//          stores left TH=RT so the 168 MB table is warm in the 192 MB L2.
// Phase 2: gather rbf rows via idx_kj (L2-resident), multiply by per-triplet
//          Legendre basis staged in LDS (async global->LDS, ASYNCcnt).
//          Output stream stored non-temporally so the 336 MB write stream
//          does not evict the rbf gather table from L2.
// Bandwidth-bound (~0.7 GB total traffic @ 23.3 TB/s); no contraction -> no WMMA.

constexpr int kNSph  = 7;
constexpr int kNRad  = 6;
constexpr int kNJ    = 42;   // kNSph * kNRad
constexpr int kBlock = 256;  // 8 wave32 waves per block

#if defined(__gfx1250__) && __has_builtin(__builtin_amdgcn_global_load_async_to_lds_b32)
#define USE_ASYNC_LDS 1
#endif

typedef __attribute__((address_space(1))) int g_int;
typedef __attribute__((address_space(3))) int l_int;

__device__ __forceinline__ void g2l_b32(const void* g, void* l) {
#ifdef USE_ASYNC_LDS
  // global_load_async_to_lds_b32 : tracked by ASYNCcnt, no VGPR round trip.
  __builtin_amdgcn_global_load_async_to_lds_b32(
      (g_int*)g, (l_int*)l, /*offset=*/0, /*cpol=*/0);
#else
  *(unsigned*)l = *(const unsigned*)g;
#endif
}

__device__ __forceinline__ void wait_g2l() {
#ifdef USE_ASYNC_LDS
#if __has_builtin(__builtin_amdgcn_s_wait_asynccnt)
  __builtin_amdgcn_s_wait_asynccnt(0);
#else
  asm volatile("s_wait_asynccnt 0" ::: "memory");
#endif
#endif
}

// ---------------------------------------------------------------------------
// Kernel 1: rbf[o] for o in [0, E*42). One thread per element -> the 168 MB
// store stream is perfectly coalesced across wave32 lanes.
// ---------------------------------------------------------------------------
__global__ __launch_bounds__(kBlock) void rbf_kernel(
    const float* __restrict__ dist,
    const float* __restrict__ zeros_lk,
    const float* __restrict__ norm_lk,
    float* __restrict__ rbf,
    unsigned nElem) {
  __shared__ float s_z[kNJ];
  __shared__ float s_n[kNJ];
  const unsigned tid = threadIdx.x;
  if (tid < (unsigned)kNJ) {
    g2l_b32(zeros_lk + tid, s_z + tid);
  } else if (tid < 2u * kNJ) {
    g2l_b32(norm_lk + (tid - kNJ), s_n + (tid - kNJ));
  }
  wait_g2l();
  __syncthreads();

  const unsigned o = blockIdx.x * kBlock + tid;
  if (o >= nElem) return;
  const unsigned e = o / (unsigned)kNJ;
  const unsigned j = o - e * (unsigned)kNJ;
  const unsigned l = j / (unsigned)kNRad;

  const float x   = dist[e] * 0.2f;          // dist / CUTOFF
  const float arg = x * s_z[j];              // arg in ~[0.06, 28] -> hw sin ok
  float s, c;
  __sincosf(arg, &s, &c);
  const float inv = 1.0f / arg;

  // spherical Bessel j_l via upward recurrence (matches reference exactly)
  const float j0 = s * inv;
  float jl = j0;
  if (l > 0u) {
    float jm = j0;
    float jc = (j0 - c) * inv;               // j1 = sin/x^2 - cos/x
    for (unsigned i = 2; i <= l; ++i) {      // <=5 masked iterations per wave
      const float nx = (float)(2 * (int)i - 1) * inv * jc - jm;
      jm = jc;
      jc = nx;
    }
    jl = jc;
  }

  // envelope, p=6: 1/x - 28 x^5 + 48 x^6 - 21 x^7
  const float x2 = x * x;
  const float x5 = x2 * x2 * x;
  const float env = 1.0f / x + x5 * (-28.0f + x * (48.0f - 21.0f * x));

  // TH=RT (default): keep the table resident in L2 for the phase-2 gather.
  rbf[o] = env * s_n[j] * jl;
}

// ---------------------------------------------------------------------------
// Kernel 2: each block owns 256 triplets. Stage angle/idx into LDS (async),
// compute cbf once per triplet into LDS, then emit 256*42 outputs with
// coalesced gather reads (42-float contiguous rows, L2-resident) and
// coalesced NON-TEMPORAL stores (don't evict the rbf table from L2).
// ---------------------------------------------------------------------------
__global__ __launch_bounds__(kBlock) void triplet_kernel(
    const float* __restrict__ angle,
    const int* __restrict__ idx_kj,
    const float* __restrict__ rbf,
    float* __restrict__ out,
    int nT_total) {
  __shared__ float s_ang[kBlock];
  __shared__ int   s_idx[kBlock];
  __shared__ float s_cbf[kBlock * kNSph];

  const int tid  = threadIdx.x;
  const int base = blockIdx.x * kBlock;
  int nT = nT_total - base;
  if (nT > kBlock) nT = kBlock;

  if (tid < nT) {
    g2l_b32(angle + base + tid, s_ang + tid);
    g2l_b32(idx_kj + base + tid, s_idx + tid);
  }
  wait_g2l();
  __syncthreads();

  if (tid < nT) {
    const float ct = __cosf(s_ang[tid]);
    // coef[l] = sqrt((2l+1)/(4*pi))
    const float coef[kNSph] = {0.28209479177387814f, 0.4886025119029199f,
                               0.6307831305050401f,  0.7463526651802308f,
                               0.8462843753216345f,  0.9356025796273889f,
                               1.0171072362820548f};
    float Pm = 1.0f, Pc = ct;
    s_cbf[tid * kNSph + 0] = coef[0];
    s_cbf[tid * kNSph + 1] = coef[1] * ct;
#pragma unroll
    for (int l = 2; l < kNSph; ++l) {
      const float Pn =
          ((float)(2 * l - 1) * ct * Pc - (float)(l - 1) * Pm) * (1.0f / (float)l);
      Pm = Pc;
      Pc = Pn;
      s_cbf[tid * kNSph + l] = coef[l] * Pc;
    }
  }
  __syncthreads();

  const unsigned total = (unsigned)nT * (unsigned)kNJ;
  const unsigned obase = (unsigned)base * (unsigned)kNJ;
  for (unsigned u = (unsigned)tid; u < total; u += kBlock) {
    const unsigned tl = u / (unsigned)kNJ;
    const unsigned jj = u - tl * (unsigned)kNJ;
    const unsigned l  = jj / (unsigned)kNRad;

    const unsigned un = u + kBlock;
    if (un < total) {  // prefetch next iteration's gather row (global_prefetch_b8)
      const unsigned tl2 = un / (unsigned)kNJ;
      __builtin_prefetch(rbf + (unsigned)s_idx[tl2] * (unsigned)kNJ, 0, 3);
    }

    const float v = rbf[(unsigned)s_idx[tl] * (unsigned)kNJ + jj] *
                    s_cbf[tl * (unsigned)kNSph + l];
    // Non-temporal: write-once stream, keep L2 for the gather table.
    __builtin_nontemporal_store(v, &out[obase + u]);
  }
}

extern "C" void kernel_launch(void* const* d_in, const int* in_sizes, int n_in,
                              void* d_out, int out_size, void* d_ws, size_t ws_size,
                              hipStream_t stream) {
  (void)n_in; (void)out_size; (void)ws_size;
  const float* dist   = (const float*)d_in[0];
  const float* angle  = (const float*)d_in[1];
  const int*   idx_kj = (const int*)d_in[2];
  const float* zeros  = (const float*)d_in[3];
  const float* norm   = (const float*)d_in[4];
  float* out = (float*)d_out;

  const int E = in_sizes[0];
  const int T = in_sizes[1];

  // Workspace: rbf table, E*42 floats (168 MB for E=1M) -> fits in 192 MB L2.
  float* rbf = (float*)d_ws;

  const unsigned nElem = (unsigned)E * (unsigned)kNJ;
  const int g1 = (int)((nElem + kBlock - 1) / kBlock);
  rbf_kernel<<<g1, kBlock, 0, stream>>>(dist, zeros, norm, rbf, nElem);

  const int g2 = (T + kBlock - 1) / kBlock;
  triplet_kernel<<<g2, kBlock, 0, stream>>>(angle, idx_kj, rbf, out, T);
}